// GATActorCritic_11708080849041
// MI455X (gfx1250) — compile-verified
//
#include <hip/hip_runtime.h>
#include <math.h>

// ---------------------------------------------------------------------------
// GATActorCritic for MI455X (gfx1250). f32 WMMA (16x16x4) for all dense GEMMs,
// atomic-based segment softmax/aggregation for the 3x1M-edge sparse passes.
// ---------------------------------------------------------------------------

typedef float v2f __attribute__((ext_vector_type(2)));
typedef float v8f __attribute__((ext_vector_type(8)));

#define HIDW 64
#define LDS_STRIDE 68   // 16-row A tile, padded to kill LDS bank conflicts

// ----------------------------- WMMA GEMM -----------------------------------
// Y[nrows,64] = X[nrows,K] @ W[K,64] (+ bias).  One wave per 16-row tile.
// A tile staged in per-wave LDS; B fragments (whole 64-wide W panel) kept in
// registers and reused across the grid-stride tile loop.
template<int K>
__global__ __launch_bounds__(256) void gemm_wmma_k(
    const float* __restrict__ X, const float* __restrict__ W,
    const float* __restrict__ bias, float* __restrict__ Y, int nrows)
{
  constexpr int KK = K / 4;
  __shared__ float lds[8][16 * LDS_STRIDE];
  const int tid  = threadIdx.x;
  const int wave = tid >> 5;
  const int lane = tid & 31;
  const int half = lane >> 4;   // 0: lanes 0-15, 1: lanes 16-31
  const int lm   = lane & 15;

  // B fragments: b = {W[k0][n], W[k0+1][n]}, k0 = kk*4 + 2*half, n = nt*16+lm
  v2f wb[KK][4];
#pragma unroll
  for (int kk = 0; kk < KK; ++kk) {
    const int k0 = kk * 4 + 2 * half;
#pragma unroll
    for (int nt = 0; nt < 4; ++nt) {
      const int n = nt * 16 + lm;
      wb[kk][nt][0] = W[k0 * HIDW + n];
      wb[kk][nt][1] = W[(k0 + 1) * HIDW + n];
    }
  }
  float bv[4];
#pragma unroll
  for (int nt = 0; nt < 4; ++nt) bv[nt] = bias ? bias[nt * 16 + lm] : 0.0f;

  float* tile = &lds[wave][0];
  const int ntiles = (nrows + 15) >> 4;
  for (int t = blockIdx.x * 8 + wave; t < ntiles; t += gridDim.x * 8) {
    const int r0 = t << 4;
    // stage 16 x K floats (float4, coalesced; zero-pad rows >= nrows)
    for (int i = lane; i < 16 * (K / 4); i += 32) {
      const int row = i / (K / 4);
      const int c4  = i % (K / 4);
      const int gr  = r0 + row;
      float4 v = make_float4(0.f, 0.f, 0.f, 0.f);
      if (gr < nrows) v = *(const float4*)(X + (size_t)gr * K + c4 * 4);
      *(float4*)(tile + row * LDS_STRIDE + c4 * 4) = v;
    }
    asm volatile("s_wait_dscnt 0" ::: "memory");  // wave-local LDS RAW

    v8f acc0 = {0}, acc1 = {0}, acc2 = {0}, acc3 = {0};
#pragma unroll
    for (int kk = 0; kk < KK; ++kk) {
      const int kb = kk * 4 + 2 * half;
      v2f a;
      a[0] = tile[lm * LDS_STRIDE + kb];
      a[1] = tile[lm * LDS_STRIDE + kb + 1];
      acc0 = __builtin_amdgcn_wmma_f32_16x16x4_f32(false, a, false, wb[kk][0], (short)0, acc0, false, false);
      acc1 = __builtin_amdgcn_wmma_f32_16x16x4_f32(false, a, false, wb[kk][1], (short)0, acc1, false, false);
      acc2 = __builtin_amdgcn_wmma_f32_16x16x4_f32(false, a, false, wb[kk][2], (short)0, acc2, false, false);
      acc3 = __builtin_amdgcn_wmma_f32_16x16x4_f32(false, a, false, wb[kk][3], (short)0, acc3, false, false);
    }
    // D layout: lane holds col n=lm, rows v + 8*half
#pragma unroll
    for (int v = 0; v < 8; ++v) {
      const int row = r0 + v + 8 * half;
      if (row < nrows) {
        float* yr = Y + (size_t)row * HIDW + lm;
        yr[0]  = acc0[v] + bv[0];
        yr[16] = acc1[v] + bv[1];
        yr[32] = acc2[v] + bv[2];
        yr[48] = acc3[v] + bv[3];
      }
    }
  }
}

// -------------------- attention-logit precompute ---------------------------
// p[h,k] = sum_c W[k, h*32+c] * a_dst[h,c]   (128 values)
__global__ void pvec_kernel(const float* __restrict__ W, const float* __restrict__ adst,
                            float* __restrict__ p)
{
  const int t = threadIdx.x;            // 128 threads
  const int h = t >> 6, k = t & 63;
  float s = 0.f;
  for (int c = 0; c < 32; ++c) s += W[k * 64 + h * 32 + c] * adst[h * 32 + c];
  p[t] = s;
}

// al_s[n,h] = xs[n,h,:] . a_src[h,:] ; al_d[n,h] = h_dst[n,:] . p[h,:]
// also initializes segment max (-inf) and denom (0).
__global__ void attn_init(const float* __restrict__ xs, const float* __restrict__ hd,
                          const float* __restrict__ asrc, const float* __restrict__ pd,
                          float* __restrict__ al_s, float* __restrict__ al_d,
                          float* __restrict__ mx, float* __restrict__ den, int n)
{
  const int gid = blockIdx.x * blockDim.x + threadIdx.x;
  if (gid >= n * 2) return;
  const int node = gid >> 1, h = gid & 1;
  const float* xr = xs + (size_t)node * 64 + h * 32;
  const float* av = asrc + h * 32;
  float s = 0.f;
  for (int c = 0; c < 32; ++c) s += xr[c] * av[c];
  const float* yr = hd + (size_t)node * 64;
  const float* pv = pd + h * 64;
  float d = 0.f;
  for (int k = 0; k < 64; ++k) d += yr[k] * pv[k];
  al_s[gid] = s;
  al_d[gid] = d;
  mx[gid] = -INFINITY;
  den[gid] = 0.f;
}

// ----------------------------- edge passes ---------------------------------
__device__ __forceinline__ float lrelu02(float x) { return x > 0.f ? x : 0.2f * x; }

__device__ __forceinline__ void atomicMaxFloat(float* addr, float v)
{
  if (__float_as_uint(v) == 0x80000000u) v = 0.0f;   // canonicalize -0
  if (v >= 0.0f) atomicMax((int*)addr, __float_as_int(v));
  else           atomicMin((unsigned int*)addr, __float_as_uint(v));
}

__global__ void edge_max(const int* __restrict__ src, const int* __restrict__ dst,
                         const float* __restrict__ al_s, const float* __restrict__ al_d,
                         float* __restrict__ mx, int E)
{
  const int e = blockIdx.x * blockDim.x + threadIdx.x;
  if (e >= E) return;
  const int s = src[e], d = dst[e];
#pragma unroll
  for (int h = 0; h < 2; ++h) {
    const float al = lrelu02(al_s[s * 2 + h] + al_d[d * 2 + h]);
    atomicMaxFloat(&mx[d * 2 + h], al);
  }
}

__global__ void edge_sum(const int* __restrict__ src, const int* __restrict__ dst,
                         const float* __restrict__ al_s, const float* __restrict__ al_d,
                         const float* __restrict__ mx, float* __restrict__ den, int E)
{
  const int e = blockIdx.x * blockDim.x + threadIdx.x;
  if (e >= E) return;
  const int s = src[e], d = dst[e];
#pragma unroll
  for (int h = 0; h < 2; ++h) {
    const float al = lrelu02(al_s[s * 2 + h] + al_d[d * 2 + h]);
    atomicAdd(&den[d * 2 + h], expf(al - mx[d * 2 + h]));
  }
}

// 64 threads per edge: out[dst, c] += xs[src, c] * softmax_weight(e, h=c/32)
__global__ void edge_aggr(const int* __restrict__ src, const int* __restrict__ dst,
                          const float* __restrict__ al_s, const float* __restrict__ al_d,
                          const float* __restrict__ mx, const float* __restrict__ den,
                          const float* __restrict__ xs, float* __restrict__ out, int E)
{
  const int gid = blockIdx.x * blockDim.x + threadIdx.x;
  const int e = gid >> 6;
  if (e >= E) return;
  const int c = gid & 63;
  const int h = c >> 5;
  const int s = src[e], d = dst[e];
  const float al = lrelu02(al_s[s * 2 + h] + al_d[d * 2 + h]);
  const float w  = expf(al - mx[d * 2 + h]) / (den[d * 2 + h] + 1e-16f);
  atomicAdd(&out[(size_t)d * 64 + c], xs[(size_t)s * 64 + c] * w);
}

// ----------------------- layer init / relu / pool --------------------------
__global__ void init_out(float* __restrict__ hz_out, float* __restrict__ ha_out,
                         const float* __restrict__ gb, int n)
{
  const int gid = blockIdx.x * blockDim.x + threadIdx.x;
  if (gid >= n * 64) return;
  const int c = gid & 63;
  hz_out[gid] = gb[c] + gb[64 + c];   // bias(zz) + bias(az)
  ha_out[gid] = gb[128 + c];          // bias(za)
}

__global__ void relu2(float* __restrict__ a, float* __restrict__ b, int total)
{
  const int gid = blockIdx.x * blockDim.x + threadIdx.x;
  if (gid >= total) return;
  a[gid] = fmaxf(a[gid], 0.f);
  b[gid] = fmaxf(b[gid], 0.f);
}

__global__ void zero_pool(float* __restrict__ p)
{
  if (threadIdx.x < 128) p[threadIdx.x] = 0.f;
}

__global__ void pool_kernel(const float* __restrict__ ha, const float* __restrict__ hz,
                            float* __restrict__ pooled, int n)
{
  __shared__ float sa[256], sz[256];
  const int t = threadIdx.x;
  const int c = t & 63;
  float aacc = 0.f, zacc = 0.f;
  for (int r = blockIdx.x * 4 + (t >> 6); r < n; r += gridDim.x * 4) {
    aacc += ha[(size_t)r * 64 + c];
    zacc += hz[(size_t)r * 64 + c];
  }
  sa[t] = aacc; sz[t] = zacc;
  __syncthreads();
  if (t < 64) {
    const float a = sa[t] + sa[t + 64] + sa[t + 128] + sa[t + 192];
    const float z = sz[t] + sz[t + 64] + sz[t + 128] + sz[t + 192];
    atomicAdd(&pooled[t], a);
    atomicAdd(&pooled[64 + t], z);
  }
}

// ------------------------------- MLP head ----------------------------------
__global__ void head_kernel(const float* __restrict__ pooled, const float* __restrict__ gvec,
                            const unsigned char* __restrict__ mask,
                            const float* __restrict__ aW1, const float* __restrict__ ab1,
                            const float* __restrict__ aW2, const float* __restrict__ ab2,
                            const float* __restrict__ cW1, const float* __restrict__ cb1,
                            const float* __restrict__ cW2, const float* __restrict__ cb2,
                            float* __restrict__ out, float invN)
{
  __shared__ float feats[134];
  __shared__ float hid[128];
  __shared__ int anym;
  const int t = threadIdx.x;          // 192 threads
  if (t < 128) feats[t] = pooled[t] * invN;
  else if (t < 134) feats[t] = gvec[t - 128];
  if (t == 0) anym = 0;
  __syncthreads();
  if (t < 64 && mask[t] != 0) atomicOr(&anym, 1);
  if (t < 64) {
    float s = ab1[t];
    for (int i = 0; i < 134; ++i) s += feats[i] * aW1[i * 64 + t];
    hid[t] = fmaxf(s, 0.f);
  } else if (t < 128) {
    const int j = t - 64;
    float s = cb1[j];
    for (int i = 0; i < 134; ++i) s += feats[i] * cW1[i * 64 + j];
    hid[64 + j] = fmaxf(s, 0.f);
  }
  __syncthreads();
  if (t < 64) {
    float s = ab2[t];
    for (int k = 0; k < 64; ++k) s += hid[k] * aW2[k * 64 + t];
    const bool keep = anym ? (mask[t] != 0) : true;
    out[t] = keep ? s : -INFINITY;
  } else if (t == 64) {
    float s = cb2[0];
    for (int k = 0; k < 64; ++k) s += hid[64 + k] * cW2[k];
    out[64] = s;
  }
}

// ------------------------------- launcher ----------------------------------
extern "C" void kernel_launch(void* const* d_in, const int* in_sizes, int n_in,
                              void* d_out, int out_size, void* d_ws, size_t ws_size,
                              hipStream_t stream)
{
  (void)n_in; (void)out_size; (void)ws_size;
  const float* x_asset  = (const float*)d_in[0];
  const float* x_zone   = (const float*)d_in[1];
  const int*   ei_zz    = (const int*)d_in[2];
  const int*   ei_az    = (const int*)d_in[3];
  const int*   ei_za    = (const int*)d_in[4];
  const float* gvec     = (const float*)d_in[5];
  const unsigned char* mask = (const unsigned char*)d_in[6];
  const float* pWa      = (const float*)d_in[7];
  const float* pba      = (const float*)d_in[8];
  const float* pWz      = (const float*)d_in[9];
  const float* pbz      = (const float*)d_in[10];
  const float* gat_W    = (const float*)d_in[11];
  const float* gat_asrc = (const float*)d_in[12];
  const float* gat_adst = (const float*)d_in[13];
  const float* gat_bias = (const float*)d_in[14];
  const float* aW1 = (const float*)d_in[15];
  const float* ab1 = (const float*)d_in[16];
  const float* aW2 = (const float*)d_in[17];
  const float* ab2 = (const float*)d_in[18];
  const float* cW1 = (const float*)d_in[19];
  const float* cb1 = (const float*)d_in[20];
  const float* cW2 = (const float*)d_in[21];
  const float* cb2 = (const float*)d_in[22];
  float* out = (float*)d_out;

  const int N = in_sizes[0] / 16;
  const int E = in_sizes[2] / 2;

  // workspace layout (floats): 5*N*64 + 4*N*2 + 256  (~131 MB, fits 192MB L2)
  float* ws = (float*)d_ws;
  const size_t NH = (size_t)N * 64, N2 = (size_t)N * 2;
  float* hzA   = ws;
  float* hzB   = hzA + NH;
  float* haA   = hzB + NH;
  float* haB   = haA + NH;
  float* xs    = haB + NH;
  float* al_s  = xs + NH;
  float* al_d  = al_s + N2;
  float* mx    = al_d + N2;
  float* den   = mx + N2;
  float* pooled = den + N2;     // 128
  float* pvecb  = pooled + 128; // 128

  int gemmBlocks = (int)(((N + 15) / 16 + 7) / 8);
  if (gemmBlocks < 1) gemmBlocks = 1;
  const int nb_ne = (int)((N2 + 255) / 256);
  const int nb_e  = (E + 255) / 256;
  const int nb_ec = (int)(((size_t)E * 64 + 255) / 256);
  const int nb_nh = (int)((NH + 255) / 256);

  // input projections (WMMA, K=16 / K=32)
  gemm_wmma_k<16><<<gemmBlocks, 256, 0, stream>>>(x_asset, pWa, pba, haA, N);
  gemm_wmma_k<32><<<gemmBlocks, 256, 0, stream>>>(x_zone,  pWz, pbz, hzA, N);

  for (int l = 0; l < 2; ++l) {
    float* hz_in  = l ? hzB : hzA;
    float* ha_in  = l ? haB : haA;
    float* hz_out = l ? hzA : hzB;
    float* ha_out = l ? haA : haB;
    init_out<<<nb_nh, 256, 0, stream>>>(hz_out, ha_out, gat_bias + l * 192, N);
    for (int et = 0; et < 3; ++et) {   // 0: zz, 1: az, 2: za
      const float* hs  = (et == 1) ? ha_in : hz_in;
      const float* hd  = (et == 2) ? ha_in : hz_in;
      float*       acc = (et == 2) ? ha_out : hz_out;
      const int*   ei  = (et == 0) ? ei_zz : (et == 1) ? ei_az : ei_za;
      const float* W    = gat_W    + (size_t)(l * 3 + et) * 64 * 64;
      const float* asrc = gat_asrc + (size_t)(l * 3 + et) * 64;
      const float* adst = gat_adst + (size_t)(l * 3 + et) * 64;

      gemm_wmma_k<64><<<gemmBlocks, 256, 0, stream>>>(hs, W, nullptr, xs, N);
      pvec_kernel<<<1, 128, 0, stream>>>(W, adst, pvecb);
      attn_init<<<nb_ne, 256, 0, stream>>>(xs, hd, asrc, pvecb, al_s, al_d, mx, den, N);
      edge_max <<<nb_e, 256, 0, stream>>>(ei, ei + E, al_s, al_d, mx, E);
      edge_sum <<<nb_e, 256, 0, stream>>>(ei, ei + E, al_s, al_d, mx, den, E);
      edge_aggr<<<nb_ec, 256, 0, stream>>>(ei, ei + E, al_s, al_d, mx, den, xs, acc, E);
    }
    relu2<<<nb_nh, 256, 0, stream>>>(hz_out, ha_out, (int)NH);
  }

  // final features live in the A buffers (l=1 wrote/relu'd there)
  zero_pool<<<1, 128, 0, stream>>>(pooled);
  pool_kernel<<<240, 256, 0, stream>>>(haA, hzA, pooled, N);
  head_kernel<<<1, 192, 0, stream>>>(pooled, gvec, mask,
                                     aW1, ab1, aW2, ab2, cW1, cb1, cW2, cb2,
                                     out, 1.0f / (float)N);
}